// BatchWiseTripletDistanceLoss_13786845020971
// MI455X (gfx1250) — compile-verified
//
#include <hip/hip_runtime.h>
#include <hip/hip_bf16.h>

#define NROWS 512
#define DIM   256
#define MARGIN 0.15f
#define EPS    1e-8f

typedef __attribute__((ext_vector_type(2))) float v2f;
typedef __attribute__((ext_vector_type(8))) float v8f;

// ---------------------------------------------------------------------------
// Kernel 1: Gram matrix G = S * S^T via V_WMMA_F32_16X16X4_F32.
// One wave (32 lanes) computes one 16x16 tile of G. 512/16 = 32 tiles/side,
// 1024 tiles total = 128 blocks x 8 waves. EXEC is all-ones per WMMA rules.
//
// f32 A-matrix 16x4 layout (ISA 7.12.2): lane L holds row m = L%16;
// VGPR0 = K = 2*(L/16), VGPR1 = K = 2*(L/16)+1. For B = S^T the per-lane
// load has the identical shape with the column-tile's rows.
// C/D 16x16 f32: VGPR r, lane L -> row = r + 8*(L/16), col = L%16.
// ---------------------------------------------------------------------------
__global__ void __launch_bounds__(256)
gram_wmma_kernel(const float* __restrict__ S, float* __restrict__ G) {
    const int wave = threadIdx.x >> 5;
    const int lane = threadIdx.x & 31;
    const int tile = blockIdx.x * 8 + wave;   // 0..1023
    const int ti = tile >> 5;                 // row tile 0..31
    const int tj = tile & 31;                 // col tile 0..31

    const int m     = lane & 15;
    const int kbase = (lane >> 4) << 1;       // 0 for lanes 0-15, 2 for 16-31

    const float* __restrict__ arow = S + (size_t)(ti * 16 + m) * DIM + kbase;
    const float* __restrict__ brow = S + (size_t)(tj * 16 + m) * DIM + kbase;

    v8f acc = {};
#pragma unroll 8
    for (int k = 0; k < DIM; k += 4) {
        v2f a = *(const v2f*)(arow + k);
        v2f b = *(const v2f*)(brow + k);
        // (neg_a, A, neg_b, B, c_mod, C, reuse_a, reuse_b)
        acc = __builtin_amdgcn_wmma_f32_16x16x4_f32(
            false, a, false, b, (short)0, acc, false, false);
    }

    const int rowBase = ti * 16 + ((lane >> 4) << 3);
    const int col     = tj * 16 + (lane & 15);
#pragma unroll
    for (int r = 0; r < 8; ++r) {
        G[(size_t)(rowBase + r) * NROWS + col] = acc[r];
    }
}

// ---------------------------------------------------------------------------
// Kernel 2: row norms from the Gram diagonal.
// ---------------------------------------------------------------------------
__global__ void norms_kernel(const float* __restrict__ G, float* __restrict__ nrm) {
    int i = blockIdx.x * blockDim.x + threadIdx.x;
    if (i < NROWS) {
        nrm[i] = sqrtf(G[(size_t)i * NROWS + i]);
    }
}

// ---------------------------------------------------------------------------
// Kernel 3: per-triplet hinge terms, deterministic block reduction.
// loss_t = relu(g_an/max(na*nn,eps) - g_ap/max(na*np,eps) + margin)
// (identical to relu(d_pos - d_neg + margin) with d = 1 - cos)
// ---------------------------------------------------------------------------
#define RED_BLOCKS 64
#define RED_THREADS 256

__global__ void __launch_bounds__(RED_THREADS)
triplet_partial_kernel(const float* __restrict__ G,
                       const float* __restrict__ nrm,
                       const int* __restrict__ Ai,
                       const int* __restrict__ Pi,
                       const int* __restrict__ Ni,
                       int T,
                       float* __restrict__ partials) {
    __shared__ float sdata[RED_THREADS];
    float sum = 0.0f;
    for (int t = blockIdx.x * blockDim.x + threadIdx.x; t < T;
         t += gridDim.x * blockDim.x) {
        const int a = Ai[t];
        const int p = Pi[t];
        const int n = Ni[t];
        const float na  = nrm[a];
        const float cap = G[(size_t)a * NROWS + p] / fmaxf(na * nrm[p], EPS);
        const float can = G[(size_t)a * NROWS + n] / fmaxf(na * nrm[n], EPS);
        sum += fmaxf(can - cap + MARGIN, 0.0f);
    }
    sdata[threadIdx.x] = sum;
    __syncthreads();
#pragma unroll
    for (int s = RED_THREADS / 2; s > 0; s >>= 1) {
        if (threadIdx.x < s) sdata[threadIdx.x] += sdata[threadIdx.x + s];
        __syncthreads();
    }
    if (threadIdx.x == 0) partials[blockIdx.x] = sdata[0];
}

// ---------------------------------------------------------------------------
// Kernel 4: deterministic final sum (fixed order), scaled by 1/T.
// ---------------------------------------------------------------------------
__global__ void finalize_kernel(const float* __restrict__ partials, int nparts,
                                float invT, float* __restrict__ out) {
    if (threadIdx.x == 0 && blockIdx.x == 0) {
        float s = 0.0f;
        for (int i = 0; i < nparts; ++i) s += partials[i];
        out[0] = s * invT;
    }
}

extern "C" void kernel_launch(void* const* d_in, const int* in_sizes, int n_in,
                              void* d_out, int out_size, void* d_ws, size_t ws_size,
                              hipStream_t stream) {
    const float* samples = (const float*)d_in[0];
    // d_in[1] = targets (unused; triplets are pre-mined on host)
    const int* anchor_idx = (const int*)d_in[2];
    const int* pos_idx    = (const int*)d_in[3];
    const int* neg_idx    = (const int*)d_in[4];
    const int  T          = in_sizes[2];

    float* G        = (float*)d_ws;                    // 512*512 floats = 1 MB
    float* nrm      = G + (size_t)NROWS * NROWS;       // 512 floats
    float* partials = nrm + NROWS;                     // 64 floats

    // 1) Gram matrix via f32 WMMA: 1024 tiles, 8 waves per block.
    gram_wmma_kernel<<<128, 256, 0, stream>>>(samples, G);

    // 2) Row norms from diagonal.
    norms_kernel<<<(NROWS + 255) / 256, 256, 0, stream>>>(G, nrm);

    // 3) Triplet hinge partial sums.
    triplet_partial_kernel<<<RED_BLOCKS, RED_THREADS, 0, stream>>>(
        G, nrm, anchor_idx, pos_idx, neg_idx, T, partials);

    // 4) Deterministic finalize: mean over T.
    finalize_kernel<<<1, 32, 0, stream>>>(partials, RED_BLOCKS,
                                          1.0f / (float)T, (float*)d_out);
}